// LightGCN_Encoder_21483426415045
// MI455X (gfx1250) — compile-verified
//
#include <hip/hip_runtime.h>
#include <hip/hip_bf16.h>

// ---------------------------------------------------------------------------
// LightGCN encoder for MI455X (gfx1250).
// 3x SpMM (COO scatter-atomic, fused sparse-dropout) + running sum + gather.
// Bandwidth-bound; working set (~167 MB) fits in the 192 MB L2.
// Final gather/scale runs through V_WMMA_F32_16X16X4_F32 as D = (0.25*I) x X
// (bit-exact: products are 0 or a power-of-two scale).
// ---------------------------------------------------------------------------

#define NUM_USER 100000
#define NUM_ITEM 50000
#define N_NODES  (NUM_USER + NUM_ITEM)     // 150000
#define DIM      64
#define NE2      4000000                   // 2 * N_EDGES (symmetric nnz)
#define BATCH    16384
#define NFLOATS  (N_NODES * DIM)           // 9,600,000 floats per node buffer
#define NF4      (NFLOATS / 4)             // 2,400,000 float4

typedef float v2f __attribute__((ext_vector_type(2)));
typedef float v8f __attribute__((ext_vector_type(8)));

#if defined(__gfx1250__) && __has_builtin(__builtin_amdgcn_wmma_f32_16x16x4_f32)
#define GATHER_WMMA 1
#else
#define GATHER_WMMA 0
#endif

// cur = acc = concat(user_emb, item_emb); nxt = 0   (float4-vectorized)
__global__ void lgcn_init_kernel(const float4* __restrict__ ue,
                                 const float4* __restrict__ ie,
                                 float4* __restrict__ cur,
                                 float4* __restrict__ acc,
                                 float4* __restrict__ nxt) {
    int i = blockIdx.x * blockDim.x + threadIdx.x;
    if (i >= NF4) return;
    const int UB = NUM_USER * DIM / 4;     // 1,600,000 (clean float4 boundary)
    float4 v = (i < UB) ? ue[i] : ie[i - UB];
    cur[i] = v;
    acc[i] = v;
    nxt[i] = make_float4(0.f, 0.f, 0.f, 0.f);
}

// One edge handled by 16 threads; each thread owns 4 consecutive dims
// -> per-edge gather/scatter is one contiguous 256B segment (coalesced).
// Dropped edges (mask==0) skip gather + all atomics (~50% of work).
__global__ void lgcn_spmm_kernel(const int* __restrict__ adj_row,
                                 const int* __restrict__ adj_col,
                                 const float* __restrict__ adj_val,
                                 const unsigned char* __restrict__ mask,
                                 const float* __restrict__ cur,
                                 float* __restrict__ nxt) {
    int tid = blockIdx.x * blockDim.x + threadIdx.x;
    int e = tid >> 4;
    if (e >= NE2) return;
    if (!mask[e]) return;                              // sparse dropout
    float v = adj_val[e] * 2.0f;                       // 1/(1-0.5) rescale
    int c = adj_col[e];
    int r = adj_row[e];
    int d = (tid & 15) * 4;
    const float4 x = *(const float4*)(cur + (long)c * DIM + d);
    float* dst = nxt + (long)r * DIM + d;
    // f32 hw atomics, result unused -> atomic-without-return (STOREcnt only)
    __hip_atomic_fetch_add(dst + 0, v * x.x, __ATOMIC_RELAXED, __HIP_MEMORY_SCOPE_AGENT);
    __hip_atomic_fetch_add(dst + 1, v * x.y, __ATOMIC_RELAXED, __HIP_MEMORY_SCOPE_AGENT);
    __hip_atomic_fetch_add(dst + 2, v * x.z, __ATOMIC_RELAXED, __HIP_MEMORY_SCOPE_AGENT);
    __hip_atomic_fetch_add(dst + 3, v * x.w, __ATOMIC_RELAXED, __HIP_MEMORY_SCOPE_AGENT);
}

// acc += nxt; also zero the retiring buffer so it can serve as next layer's
// accumulation target (fuses the zero pass, saves one 38 MB sweep).
__global__ void lgcn_accum_zero_kernel(float4* __restrict__ acc,
                                       const float4* __restrict__ nxt,
                                       float4* __restrict__ zbuf) {
    int i = blockIdx.x * blockDim.x + threadIdx.x;
    if (i >= NF4) return;
    float4 a = acc[i];
    float4 n = nxt[i];
    acc[i]  = make_float4(a.x + n.x, a.y + n.y, a.z + n.z, a.w + n.w);
    zbuf[i] = make_float4(0.f, 0.f, 0.f, 0.f);
}

// out[orow] = acc[node(orow)] * 0.25.  One wave per 16x16 f32 tile.
// The user/item split is tile-aligned (16384 % 16 == 0), so the select of
// index array / node offset is WAVE-UNIFORM -> no saveexec divergence around
// the index loads, and EXEC stays all-1s for the WMMA ops.
//
// WMMA path: D = sum_k A_k(16x4) * B_k(4x16), A = 0.25*I16 sliced along K.
//   A layout (32b 16x4): lanes 0-15 M=0-15 {VGPR0:K=0, VGPR1:K=1},
//                        lanes 16-31 M=0-15 {VGPR0:K=2, VGPR1:K=3}.
//   B layout (32b 4x16): lanes 0-15 N=0-15 {VGPR0:K=0, VGPR1:K=1},
//                        lanes 16-31 N=0-15 {VGPR0:K=2, VGPR1:K=3}.
//   D layout (32b 16x16): lanes 0-15 {VGPRv: M=v}, lanes 16-31 {VGPRv: M=v+8}.
__global__ void lgcn_gather_kernel(const float* __restrict__ acc,
                                   const int* __restrict__ users,
                                   const int* __restrict__ items,
                                   float* __restrict__ out) {
    const int wave = (blockIdx.x * blockDim.x + threadIdx.x) >> 5;
    const int lane = threadIdx.x & 31;
    const int row_tile = wave >> 2;          // 0..2047 (32768 rows / 16)
    const int col_tile = wave & 3;           // 0..3    (64 dims / 16)
    const int base_row = row_tile * 16;
    const int col0 = col_tile * 16;

    // wave-uniform user/item selection
    const bool is_item = (base_row >= BATCH);
    const int* __restrict__ idx = is_item ? items : users;
    const int  ibase = is_item ? (base_row - BATCH) : base_row;
    const int  nadd  = is_item ? NUM_USER : 0;

#if GATHER_WMMA
    const int m   = lane & 15;               // N-index / A-row index
    const int kk0 = (lane < 16) ? 0 : 2;     // which K-pair this half holds
    v8f c = {};
#pragma unroll
    for (int k = 0; k < 4; ++k) {
        const int ra = 4 * k + kk0;          // tile-row feeding b.x / a.x
        const int rb = ra + 1;               // tile-row feeding b.y / a.y
        const long na = (long)(idx[ibase + ra] + nadd);
        const long nb = (long)(idx[ibase + rb] + nadd);
        v2f b;
        b.x = acc[na * DIM + col0 + m];
        b.y = acc[nb * DIM + col0 + m];
        v2f a;
        a.x = (m == ra) ? 0.25f : 0.0f;      // 0.25 * I16, K-slice k
        a.y = (m == rb) ? 0.25f : 0.0f;
        c = __builtin_amdgcn_wmma_f32_16x16x4_f32(
                /*neg_a=*/false, a, /*neg_b=*/false, b,
                /*c_mod=*/(short)0, c, /*reuse_a=*/false, /*reuse_b=*/false);
    }
    const int mbase = (lane < 16) ? 0 : 8;
#pragma unroll
    for (int v = 0; v < 8; ++v) {
        const int orow = base_row + mbase + v;
        out[(long)orow * DIM + col0 + m] = c[v];
    }
#else
    for (int t = lane; t < 256; t += 32) {
        const int r = t >> 4;
        const int cc = t & 15;
        const long n = (long)(idx[ibase + r] + nadd);
        out[(long)(base_row + r) * DIM + col0 + cc] =
            acc[n * DIM + col0 + cc] * 0.25f;
    }
#endif
}

extern "C" void kernel_launch(void* const* d_in, const int* in_sizes, int n_in,
                              void* d_out, int out_size, void* d_ws, size_t ws_size,
                              hipStream_t stream) {
    const float* user_emb        = (const float*)d_in[0];
    const float* item_emb        = (const float*)d_in[1];
    const float* adj_val         = (const float*)d_in[2];
    const int*   adj_row         = (const int*)d_in[3];
    const int*   adj_col         = (const int*)d_in[4];
    const unsigned char* mask    = (const unsigned char*)d_in[5];
    const int*   users           = (const int*)d_in[6];
    const int*   items           = (const int*)d_in[7];
    float*       out             = (float*)d_out;

    // workspace: three node buffers, 38.4 MB each (256B-aligned slabs)
    float* cur = (float*)d_ws;
    float* nxt = cur + NFLOATS;
    float* acc = nxt + NFLOATS;

    const dim3 blk(256);
    lgcn_init_kernel<<<(NF4 + 255) / 256, blk, 0, stream>>>(
        (const float4*)user_emb, (const float4*)item_emb,
        (float4*)cur, (float4*)acc, (float4*)nxt);

    for (int l = 0; l < 3; ++l) {
        lgcn_spmm_kernel<<<(NE2 * 16 + 255) / 256, blk, 0, stream>>>(
            adj_row, adj_col, adj_val, mask, cur, nxt);
        lgcn_accum_zero_kernel<<<(NF4 + 255) / 256, blk, 0, stream>>>(
            (float4*)acc, (const float4*)nxt, (float4*)cur);
        float* t = cur; cur = nxt; nxt = t;   // deterministic pointer swap
    }

    // 32768 output rows -> 2048 row-tiles x 4 col-tiles = 8192 waves
    lgcn_gather_kernel<<<8192 * 32 / 256, blk, 0, stream>>>(acc, users, items, out);
}